// VI_mass_63410897158739
// MI455X (gfx1250) — compile-verified
//
#include <hip/hip_runtime.h>
#include <hip/hip_bf16.h>

#define NN 1024      // data points
#define NB 32        // Cholesky panel width
#define MC 64        // MC samples
#define THREADS 512  // 16 waves of 32
#define NWAVES (THREADS / 32)

typedef __attribute__((ext_vector_type(2))) float v2f;
typedef __attribute__((ext_vector_type(8))) float v8f;

__device__ __forceinline__ float softplusf(float v) {
    // jax.nn.softplus + 1e-7, overflow-safe
    return fmaxf(v, 0.0f) + log1pf(expf(-fabsf(v))) + 1e-7f;
}

__device__ __forceinline__ float log_normal_f(float v, float mean, float s) {
    const float C = -0.91893853320467274178f; // -0.5*log(2*pi)
    float d = v - mean;
    return -d * d / (2.0f * s * s + 1e-5f) - logf(s) + C;
}

__device__ __forceinline__ float log_lognormal_f(float v, float mean, float s) {
    const float C = -0.91893853320467274178f;
    float lv = logf(v);
    float d = lv - mean;
    return -d * d / (2.0f * s * s + 1e-6f) - lv - logf(s) + C;
}

// ---------------------------------------------------------------------------
// Kernel 1: reparameterized samples + (log_prior - log_q), one thread per m
// ---------------------------------------------------------------------------
__global__ void vi_scalars_kernel(const float* __restrict__ z,
                                  const float* qbm, const float* qbs,
                                  const float* qsm, const float* qss,
                                  const float* qem, const float* qes,
                                  const float* qzm, const float* qzs,
                                  const float* qnm, const float* qns,
                                  float* sigma2, float* beta, float* eta,
                                  float* lsZ, float* lsN, float* pq) {
    int m = blockIdx.x * blockDim.x + threadIdx.x;
    if (m >= MC) return;
    float sp_b = softplusf(qbs[0]);
    float sp_s = softplusf(qss[0]);
    float sp_e = softplusf(qes[0]);
    float sp_z = softplusf(qzs[0]);
    float sp_n = softplusf(qns[0]);

    float s2  = expf(z[m * 5 + 0] * sp_s + qsm[0]);
    float b   =      z[m * 5 + 1] * sp_b + qbm[0];
    float e   = expf(z[m * 5 + 2] * sp_e + qem[0]);
    float lz  = expf(z[m * 5 + 3] * sp_z + qzm[0]);
    float ln_ = expf(z[m * 5 + 4] * sp_n + qnm[0]);

    float s0 = softplusf(sqrtf(logf(2.0f)));  // default prior scale

    float lp = log_normal_f(b, 0.0f, 1.0f)
             + log_lognormal_f(s2, 1.0f, s0)
             + log_lognormal_f(e,  1.0f, s0)
             + log_lognormal_f(lz, 1.0f, s0)
             + log_lognormal_f(ln_, 1.0f, s0);
    float lq = log_normal_f(b, qbm[0], sp_b)
             + log_lognormal_f(s2, qsm[0], sp_s)
             + log_lognormal_f(e,  qem[0], sp_e)
             + log_lognormal_f(lz, qzm[0], sp_z)
             + log_lognormal_f(ln_, qnm[0], sp_n);

    sigma2[m] = s2; beta[m] = b; eta[m] = e; lsZ[m] = lz; lsN[m] = ln_;
    pq[m] = lp - lq;
}

// ---------------------------------------------------------------------------
// Kernel 2: dense RBF covariance per MC sample (bandwidth-trivial)
// ---------------------------------------------------------------------------
__global__ __launch_bounds__(256)
void vi_cov_kernel(const float* __restrict__ x,
                   const float* __restrict__ sigma2, const float* __restrict__ eta,
                   const float* __restrict__ lsZ, const float* __restrict__ lsN,
                   float* __restrict__ mats) {
    int m = blockIdx.z;
    int i = blockIdx.y * 16 + threadIdx.y;
    int j = blockIdx.x * 16 + threadIdx.x;
    float ilz = 1.0f / lsZ[m];
    float iln = 1.0f / lsN[m];
    float xi0 = x[i * 2 + 0] * ilz, xi1 = x[i * 2 + 1] * iln;
    float xj0 = x[j * 2 + 0] * ilz, xj1 = x[j * 2 + 1] * iln;
    float d0 = xi0 - xj0, d1 = xi1 - xj1;
    float sqd = fmaxf(d0 * d0 + d1 * d1, 0.0f);   // == n2i + n2j - 2*dot, clamped
    float v = eta[m] * expf(-0.5f * sqd);
    if (i == j) v += sigma2[m];
    mats[(size_t)m * NN * NN + (size_t)i * NN + j] = v;
}

// ---------------------------------------------------------------------------
// Kernel 3: per-sample blocked Cholesky (WMMA fp32 SYRK trailing update),
//           then blocked forward substitution, quad/logdet, final output.
//           One 512-thread (16-wave) workgroup per MC sample.
// ---------------------------------------------------------------------------
__global__ __launch_bounds__(THREADS)
void vi_chol_kernel(float* __restrict__ mats, const float* __restrict__ y,
                    const float* __restrict__ beta, const float* __restrict__ pq,
                    float* __restrict__ out) {
    __shared__ float Dg[NB][NB + 1];   // diagonal block (padded vs bank conflicts)
    __shared__ float sol[NN];          // rhs / solution vector
    __shared__ float red[THREADS];     // reduction scratch

    float* M = mats + (size_t)blockIdx.x * NN * NN;
    const int tid  = threadIdx.x;
    const int lane = tid & 31;
    const int wave = tid >> 5;

    // ---------------- blocked Cholesky, right-looking ----------------
    for (int k = 0; k < NN / NB; ++k) {
        const int koff = k * NB;

        // load diagonal block into LDS
        for (int idx = tid; idx < NB * NB; idx += THREADS) {
            int r = idx / NB, c = idx % NB;
            Dg[r][c] = M[(size_t)(koff + r) * NN + koff + c];
        }
        __syncthreads();

        // unblocked Cholesky of the 32x32 diagonal block (cooperative)
        for (int j = 0; j < NB; ++j) {
            if (tid == 0) Dg[j][j] = sqrtf(fmaxf(Dg[j][j], 1e-20f));
            __syncthreads();
            float djj = Dg[j][j];
            if (tid > j && tid < NB) Dg[tid][j] = Dg[tid][j] / djj;
            __syncthreads();
            for (int idx = tid; idx < NB * NB; idx += THREADS) {
                int r = idx / NB, c = idx % NB;
                if (c > j && r >= c) Dg[r][c] -= Dg[r][j] * Dg[c][j];
            }
            __syncthreads();
        }
        // write lower triangle of the factored diagonal block back
        for (int idx = tid; idx < NB * NB; idx += THREADS) {
            int r = idx / NB, c = idx % NB;
            if (c <= r) M[(size_t)(koff + r) * NN + koff + c] = Dg[r][c];
        }
        __syncthreads();

        const int t0 = koff + NB;          // start of trailing submatrix

        // TRSM: panel rows X solve X * L_kk^T = A ; one row per thread
        for (int r = t0 + tid; r < NN; r += THREADS) {
            float a[NB];
            for (int c = 0; c < NB; ++c) a[c] = M[(size_t)r * NN + koff + c];
            for (int c = 0; c < NB; ++c) {
                float v = a[c];
                for (int c2 = 0; c2 < c; ++c2) v -= a[c2] * Dg[c][c2];
                a[c] = v / Dg[c][c];
            }
            for (int c = 0; c < NB; ++c) M[(size_t)r * NN + koff + c] = a[c];
        }
        __syncthreads();

        // SYRK trailing update: C -= P * P^T via V_WMMA_F32_16X16X4_F32.
        // Tile-row per wave: A operand loaded once per ti, B double-buffered
        // so its 8-load clause overlaps the previous tile's WMMA chain.
        const int tn = NN - t0;
        if (tn > 0) {
            const int ntr  = tn / 16;        // 16-row tile count
            const int mrow = lane & 15;      // M/N index within tile
            const int kb   = (lane >> 4) * 2;// K sub-offset: 0 or 2
            for (int ti = wave; ti < ntr; ti += NWAVES) {   // wave-uniform
                const size_t baseA = (size_t)(t0 + ti * 16 + mrow) * NN + koff + kb;
                v2f a[8];
#pragma unroll
                for (int c8 = 0; c8 < 8; ++c8) {
                    a[c8].x = M[baseA + c8 * 4 + 0];
                    a[c8].y = M[baseA + c8 * 4 + 1];
                }
                // prologue: load B for tj = 0
                size_t baseB = (size_t)(t0 + mrow) * NN + koff + kb;
                v2f bcur[8];
#pragma unroll
                for (int c8 = 0; c8 < 8; ++c8) {
                    bcur[c8].x = M[baseB + c8 * 4 + 0];
                    bcur[c8].y = M[baseB + c8 * 4 + 1];
                }
                for (int tj = 0; tj <= ti; ++tj) {
                    v2f bnext[8];
                    if (tj < ti) {
                        size_t nb = (size_t)(t0 + (tj + 1) * 16 + mrow) * NN + koff + kb;
                        __builtin_prefetch(&M[nb + 128], 0, 1);
#pragma unroll
                        for (int c8 = 0; c8 < 8; ++c8) {
                            bnext[c8].x = M[nb + c8 * 4 + 0];
                            bnext[c8].y = M[nb + c8 * 4 + 1];
                        }
                    }
                    v8f acc = {};
#pragma unroll
                    for (int c8 = 0; c8 < 8; ++c8) {
                        acc = __builtin_amdgcn_wmma_f32_16x16x4_f32(
                            false, a[c8], false, bcur[c8], (short)0, acc, false, false);
                    }
                    // C/D layout: VGPR v -> row v + (lane/16)*8, col lane%16
                    const int crow = t0 + ti * 16 + (lane >> 4) * 8;
                    const int ccol = t0 + tj * 16 + (lane & 15);
#pragma unroll
                    for (int v = 0; v < 8; ++v) {
                        size_t off = (size_t)(crow + v) * NN + ccol;
                        M[off] -= acc[v];
                    }
                    if (tj < ti) {
#pragma unroll
                        for (int c8 = 0; c8 < 8; ++c8) bcur[c8] = bnext[c8];
                    }
                }
            }
        }
        __syncthreads();
    }

    // ---------------- forward substitution: L * sol = y - beta ----------------
    float bm = beta[blockIdx.x];
    for (int i = tid; i < NN; i += THREADS) sol[i] = y[i] - bm;
    __syncthreads();

    for (int b = 0; b < NN / NB; ++b) {
        const int boff = b * NB;
        for (int idx = tid; idx < NB * NB; idx += THREADS) {
            int r = idx / NB, c = idx % NB;
            Dg[r][c] = M[(size_t)(boff + r) * NN + boff + c];
        }
        __syncthreads();
        if (tid == 0) {
            for (int r = 0; r < NB; ++r) {
                float v = sol[boff + r];
                for (int c = 0; c < r; ++c) v -= Dg[r][c] * sol[boff + c];
                sol[boff + r] = v / Dg[r][r];
            }
        }
        __syncthreads();
        for (int r = boff + NB + tid; r < NN; r += THREADS) {
            float v = sol[r];
            for (int c = 0; c < NB; ++c) v -= M[(size_t)r * NN + boff + c] * sol[boff + c];
            sol[r] = v;
        }
        __syncthreads();
    }

    // quad = sum(sol^2); logdet = 2*sum(log(diag(L)))
    float part = 0.0f, ldp = 0.0f;
    for (int i = tid; i < NN; i += THREADS) {
        part += sol[i] * sol[i];
        ldp  += logf(M[(size_t)i * NN + i]);
    }
    red[tid] = part; __syncthreads();
    for (int s = THREADS / 2; s > 0; s >>= 1) { if (tid < s) red[tid] += red[tid + s]; __syncthreads(); }
    float quad = red[0]; __syncthreads();
    red[tid] = ldp; __syncthreads();
    for (int s = THREADS / 2; s > 0; s >>= 1) { if (tid < s) red[tid] += red[tid + s]; __syncthreads(); }
    float logdet = 2.0f * red[0];

    if (tid == 0) {
        const float LOG2PI = 1.83787706640934548356f;
        float logL = -0.5f * (quad + logdet + (float)NN * LOG2PI);
        out[blockIdx.x] = logL + pq[blockIdx.x];
    }
}

// ---------------------------------------------------------------------------
extern "C" void kernel_launch(void* const* d_in, const int* in_sizes, int n_in,
                              void* d_out, int out_size, void* d_ws, size_t ws_size,
                              hipStream_t stream) {
    const float* x   = (const float*)d_in[0];
    const float* y   = (const float*)d_in[1];
    const float* z   = (const float*)d_in[2];
    const float* qbm = (const float*)d_in[3];
    const float* qbs = (const float*)d_in[4];
    const float* qsm = (const float*)d_in[5];
    const float* qss = (const float*)d_in[6];
    const float* qem = (const float*)d_in[7];
    const float* qes = (const float*)d_in[8];
    const float* qzm = (const float*)d_in[9];
    const float* qzs = (const float*)d_in[10];
    const float* qnm = (const float*)d_in[11];
    const float* qns = (const float*)d_in[12];
    float* out = (float*)d_out;

    float* ws     = (float*)d_ws;
    float* mats   = ws;                                  // MC * NN * NN floats
    float* sigma2 = ws + (size_t)MC * NN * NN;
    float* beta   = sigma2 + MC;
    float* eta    = beta + MC;
    float* lsZ    = eta + MC;
    float* lsN    = lsZ + MC;
    float* pq     = lsN + MC;

    vi_scalars_kernel<<<1, MC, 0, stream>>>(z, qbm, qbs, qsm, qss, qem, qes,
                                            qzm, qzs, qnm, qns,
                                            sigma2, beta, eta, lsZ, lsN, pq);

    vi_cov_kernel<<<dim3(NN / 16, NN / 16, MC), dim3(16, 16), 0, stream>>>(
        x, sigma2, eta, lsZ, lsN, mats);

    vi_chol_kernel<<<MC, THREADS, 0, stream>>>(mats, y, beta, pq, out);
}